// NaiveDemosaicking_52785148068083
// MI455X (gfx1250) — compile-verified
//
#include <hip/hip_runtime.h>
#include <cstdint>

#define NT    256
#define TILE  64
#define HALO  7
#define S     (TILE + 2*HALO)     // 78
#define REGN  (S*S)               // 6084
#define IMG   1024
#define IMGSZ (IMG*IMG)

// ---- symmetric ("reflect") boundary: pad[-1]=x[0], pad[N]=x[N-1] ----
__device__ __forceinline__ int refl(int g) {
  g = (g < 0) ? (-1 - g) : g;
  g = (g >= IMG) ? (2 * IMG - 1 - g) : g;
  return g;
}

// Low 32 bits of a flat pointer into LDS == wave-relative LDS byte address.
__device__ __forceinline__ unsigned lds_off(const void* p) {
  return (unsigned)(unsigned long long)p;
}

// ---- CDNA5 async global<->LDS copies (tracked by ASYNCcnt) ----
__device__ __forceinline__ void async_load_b32(unsigned lds, unsigned voff, const void* base) {
  asm volatile("global_load_async_to_lds_b32 %0, %1, %2"
               :: "v"(lds), "v"(voff), "s"(base) : "memory");
}
__device__ __forceinline__ void async_store_b32(unsigned voff, unsigned lds, void* base) {
  asm volatile("global_store_async_from_lds_b32 %0, %1, %2"
               :: "v"(voff), "v"(lds), "s"(base) : "memory");
}
__device__ __forceinline__ void wait_async0() {
#if __has_builtin(__builtin_amdgcn_s_wait_asynccnt)
  __builtin_amdgcn_s_wait_asynccnt(0);
#else
  asm volatile("s_wait_asynccnt 0" ::: "memory");
#endif
}

// ---- selection networks ----
__device__ __forceinline__ void ce(float& a, float& b) {
  float lo = fminf(a, b), hi = fmaxf(a, b);
  a = lo; b = hi;
}
// 5th smallest of 8 (== jnp.sort(...)[4]): sort two 4-groups (any partition),
// bitonic cross stage, min of upper half.
__device__ __forceinline__ float med8(float a0, float a1, float a2, float a3,
                                      float b0, float b1, float b2, float b3) {
  ce(a0, a1); ce(a2, a3); ce(a0, a2); ce(a1, a3); ce(a1, a2);
  ce(b0, b1); ce(b2, b3); ce(b0, b2); ce(b1, b3); ce(b1, b2);
  float m0 = fmaxf(a0, b3), m1 = fmaxf(a1, b2);
  float m2 = fmaxf(a2, b1), m3 = fmaxf(a3, b0);
  return fminf(fminf(m0, m1), fminf(m2, m3));
}
// 3rd smallest of 4 (== jnp.sort(...)[2])
__device__ __forceinline__ float med4(float a, float b, float c, float d) {
  float l1 = fminf(a, b), h1 = fmaxf(a, b);
  float l2 = fminf(c, d), h2 = fmaxf(c, d);
  return fmaxf(fmaxf(l1, l2), fminf(h1, h2));
}

// ---- LDS stencil phases; H = halo of the region being WRITTEN ----
// All widths are even; each thread handles 2 horizontally adjacent pixels so
// neighbor rows are shared and adjacent dword loads combine to ds_load_2addr.
template <int H>
__device__ __forceinline__ void diffP(float* D, const float* X, const float* Y, int tid) {
  constexpr int W = TILE + 2 * H, O = HALO - H, NP = (W / 2) * W;
  for (int j = tid; j < NP; j += NT) {
    int p = (O + j / (W / 2)) * S + (O + (j % (W / 2)) * 2);
    float x0 = X[p], x1 = X[p + 1], y0 = Y[p], y1 = Y[p + 1];
    D[p] = x0 - y0;
    D[p + 1] = x1 - y1;
  }
}
template <int H>
__device__ __forceinline__ void medRB(const float* D, const float* Gc, float* X, int tid) {
  constexpr int W = TILE + 2 * H, O = HALO - H, NP = (W / 2) * W;
  for (int j = tid; j < NP; j += NT) {
    int p = (O + j / (W / 2)) * S + (O + (j % (W / 2)) * 2);
    // 12 shared loads cover both pixels' 8-neighborhoods
    float t0 = D[p - S - 1], t1 = D[p - S], t2 = D[p - S + 1], t3 = D[p - S + 2];
    float c0 = D[p - 1],     c1 = D[p],     c2 = D[p + 1],     c3 = D[p + 2];
    float b0 = D[p + S - 1], b1 = D[p + S], b2 = D[p + S + 1], b3 = D[p + S + 2];
    float g0 = Gc[p], g1 = Gc[p + 1];
    float m0 = med8(t0, t1, t2, c0, c2, b0, b1, b2);
    float m1 = med8(t1, t2, t3, c1, c3, b1, b2, b3);
    X[p] = m0 + g0;
    X[p + 1] = m1 + g1;
  }
}
template <int H>
__device__ __forceinline__ void updG(float* Gc, const float* D1, const float* D2,
                                     const float* R, const float* B, int tid) {
  constexpr int W = TILE + 2 * H, O = HALO - H, NP = (W / 2) * W;
  for (int j = tid; j < NP; j += NT) {
    int p = (O + j / (W / 2)) * S + (O + (j % (W / 2)) * 2);
    float a_t0 = D1[p - S], a_t1 = D1[p - S + 1];
    float a_c0 = D1[p - 1], a_c1 = D1[p], a_c2 = D1[p + 1], a_c3 = D1[p + 2];
    float a_b0 = D1[p + S], a_b1 = D1[p + S + 1];
    float b_t0 = D2[p - S], b_t1 = D2[p - S + 1];
    float b_c0 = D2[p - 1], b_c1 = D2[p], b_c2 = D2[p + 1], b_c3 = D2[p + 2];
    float b_b0 = D2[p + S], b_b1 = D2[p + S + 1];
    float r0 = R[p], r1 = R[p + 1], bb0 = B[p], bb1 = B[p + 1];
    float m10 = med4(a_t0, a_c0, a_c2, a_b0);
    float m11 = med4(a_t1, a_c1, a_c3, a_b1);
    float m20 = med4(b_t0, b_c0, b_c2, b_b0);
    float m21 = med4(b_t1, b_c1, b_c3, b_b1);
    Gc[p]     = 0.5f * (m10 + m20 + r0 + bb0);
    Gc[p + 1] = 0.5f * (m11 + m21 + r1 + bb1);
  }
}
// One refinement iteration. Input fields valid at halo HIN; outputs: R,B at HIN-1, G at HIN-2.
template <int HIN>
__device__ __forceinline__ void iterStep(float* D1, float* D2, float* R, float* G, float* B, int tid) {
  diffP<HIN>(D1, R, G, tid);        __syncthreads();
  medRB<HIN - 1>(D1, G, R, tid);    __syncthreads();
  diffP<HIN>(D1, B, G, tid);        __syncthreads();
  medRB<HIN - 1>(D1, G, B, tid);    __syncthreads();
  diffP<HIN - 1>(D1, G, R, tid);
  diffP<HIN - 1>(D2, G, B, tid);    __syncthreads();
  updG<HIN - 2>(G, D1, D2, R, B, tid); __syncthreads();
}

// Demosaick accumulation for one pixel from a 3x4 tap window (CX = column shift 0/1).
template <int CX>
__device__ __forceinline__ void demo1(const float v[3][4], const int py[3], const int px[4],
                                      float& accR, float& accG, float& accB) {
  constexpr float WRB[3][3] = {{0.25f, 0.5f, 0.25f}, {0.5f, 1.0f, 0.5f}, {0.25f, 0.5f, 0.25f}};
  constexpr float WGK[3][3] = {{0.0f, 0.25f, 0.0f}, {0.25f, 1.0f, 0.25f}, {0.0f, 0.25f, 0.0f}};
  accR = 0.f; accG = 0.f; accB = 0.f;
#pragma unroll
  for (int dy = 0; dy < 3; ++dy) {
#pragma unroll
    for (int dx = 0; dx < 3; ++dx) {
      float val = v[dy][dx + CX];
      int pyy = py[dy], pxx = px[dx + CX];
      // GRBG: R at (even row, odd col); B at (odd row, even col); G where parities match
      bool isr = (pyy == 0) && (pxx == 1);
      bool isb = (pyy == 1) && (pxx == 0);
      bool isg = (pyy == pxx);
      accR += WRB[dy][dx] * (isr ? val : 0.f);
      accB += WRB[dy][dx] * (isb ? val : 0.f);
      if (WGK[dy][dx] != 0.f) accG += WGK[dy][dx] * (isg ? val : 0.f);
    }
  }
}

__global__ void __launch_bounds__(NT)
demosaick_median_fused(const float* __restrict__ in, float* __restrict__ out) {
  __shared__ float sD1[REGN];  // mosaic staging -> D scratch
  __shared__ float sR[REGN];
  __shared__ float sG[REGN];
  __shared__ float sB[REGN];
  __shared__ float sD2[REGN];

  const int tid = threadIdx.x;
  const int bx0 = blockIdx.x * TILE;
  const int by0 = blockIdx.y * TILE;
  const int bat = blockIdx.z;
  const float* src = in + (size_t)bat * IMGSZ;

  // ---- stage reflected mosaic tile+halo into LDS via async copy engine ----
  for (int i = tid; i < REGN; i += NT) {
    int ey = i / S, ex = i % S;
    int ry = refl(by0 + ey - HALO);
    int rx = refl(bx0 + ex - HALO);
    async_load_b32(lds_off(&sD1[i]), (unsigned)((ry * IMG + rx) * 4), src);
  }
  wait_async0();
  __syncthreads();

  // ---- demosaick (GRBG bilinear), valid at halo 6; classify taps by REFLECTED parity ----
  {
    constexpr int H = 6, W = TILE + 2 * H, O = HALO - H, NP = (W / 2) * W;
    for (int j = tid; j < NP; j += NT) {
      int ey = O + j / (W / 2);
      int ex = O + (j % (W / 2)) * 2;
      int py[3], px[4];
#pragma unroll
      for (int k = 0; k < 3; ++k) py[k] = refl(by0 + ey - HALO + (k - 1)) & 1;
#pragma unroll
      for (int k = 0; k < 4; ++k) px[k] = refl(bx0 + ex - HALO + (k - 1)) & 1;
      float v[3][4];
#pragma unroll
      for (int dy = 0; dy < 3; ++dy)
#pragma unroll
        for (int dx = 0; dx < 4; ++dx)
          v[dy][dx] = sD1[(ey + dy - 1) * S + (ex + dx - 1)];
      float r0, g0, b0, r1, g1, b1;
      demo1<0>(v, py, px, r0, g0, b0);
      demo1<1>(v, py, px, r1, g1, b1);
      int p = ey * S + ex;
      sR[p] = r0; sG[p] = g0; sB[p] = b0;
      sR[p + 1] = r1; sG[p + 1] = g1; sB[p + 1] = b1;
    }
  }
  __syncthreads();

  // ---- 3 median refinement iterations (halo 6 -> 4 -> 2 -> 0) ----
  iterStep<6>(sD1, sD2, sR, sG, sB, tid);
  iterStep<4>(sD1, sD2, sR, sG, sB, tid);
  iterStep<2>(sD1, sD2, sR, sG, sB, tid);

  // ---- stream R,G,B planes from LDS straight to global via async stores ----
  float* dR = out + ((size_t)bat * 3 + 0) * IMGSZ;
  float* dG = out + ((size_t)bat * 3 + 1) * IMGSZ;
  float* dB = out + ((size_t)bat * 3 + 2) * IMGSZ;
  for (int j = tid; j < (TILE / 2) * TILE; j += NT) {
    int y = j >> 5, x = (j & 31) * 2;
    unsigned voff = (unsigned)((((by0 + y) * IMG) + bx0 + x) * 4);
    int p = (HALO + y) * S + (HALO + x);
    async_store_b32(voff,     lds_off(&sR[p]),     dR);
    async_store_b32(voff + 4, lds_off(&sR[p + 1]), dR);
    async_store_b32(voff,     lds_off(&sG[p]),     dG);
    async_store_b32(voff + 4, lds_off(&sG[p + 1]), dG);
    async_store_b32(voff,     lds_off(&sB[p]),     dB);
    async_store_b32(voff + 4, lds_off(&sB[p + 1]), dB);
  }
  wait_async0();
}

extern "C" void kernel_launch(void* const* d_in, const int* in_sizes, int n_in,
                              void* d_out, int out_size, void* d_ws, size_t ws_size,
                              hipStream_t stream) {
  (void)n_in; (void)out_size; (void)d_ws; (void)ws_size;
  const float* in = (const float*)d_in[0];
  float* out = (float*)d_out;
  int nb = in_sizes[0] / IMGSZ;                 // 8 batches of 1024x1024
  dim3 grid(IMG / TILE, IMG / TILE, nb);        // 16 x 16 x 8 = 2048 workgroups
  demosaick_median_fused<<<grid, dim3(NT), 0, stream>>>(in, out);
}